// AnchorTarget_20933670600880
// MI455X (gfx1250) — compile-verified
//
#include <hip/hip_runtime.h>
#include <hip/hip_bf16.h>

typedef unsigned int u32;
typedef __attribute__((ext_vector_type(4))) u32 u32x4;
typedef __attribute__((ext_vector_type(8))) u32 u32x8;
typedef __attribute__((ext_vector_type(4))) float f32x4;

// 9 base anchors for STRIDE=8, ANCHOR_SIZE=32, ratios {0.5,1,2}, scales {1, 2^(1/3), 2^(2/3)}.
// Computed in float64 exactly as the numpy reference (np.round is banker's: round(22.5)=22).
__constant__ float c_base[9][4] = {
    {-6.5f,                 5.0f,                 37.5f,                26.0f               },
    {-12.348223622634647f,  2.140868451156394f,   43.348223622634647f,  28.859131548843606f },
    {-19.716523669284486f, -1.461411571650194f,   50.716523669284486f,  32.461411571650194f },
    { 0.0f,                 0.0f,                 31.0f,                31.0f               },
    {-4.158736798317972f,  -4.158736798317972f,   35.158736798317972f,  35.158736798317972f },
    {-9.398416831491190f,  -9.398416831491190f,   40.398416831491190f,  40.398416831491190f },
    { 4.5f,                -7.0f,                 26.5f,                38.0f               },
    { 1.510907926208958f, -12.978184147582083f,   29.489092073791042f,  43.978184147582083f },
    {-2.255112097634293f, -20.510224195268586f,   33.255112097634293f,  51.510224195268586f },
};

__global__ __launch_bounds__(256)
void anchor_target_kernel(const float* __restrict__ gt,
                          const int* __restrict__ pfw,
                          const int* __restrict__ pfh,
                          float* __restrict__ out,
                          int A, int G) {
    __shared__ __align__(16) float sgt[640];   // up to 128 GT rows of 5 floats

    // Generic pointer to LDS: low 32 bits are the DS byte offset (addr = {aperture, off}).
    float* sp = &sgt[0];
    u32 lds_off = (u32)(uintptr_t)sp;

    // ---- Stage GT boxes (5*G dwords) into LDS with the Tensor Data Mover ----
    // Wave 0 of the block issues one TDM descriptor; TENSORcnt tracks completion.
    if (threadIdx.x < 32) {
        unsigned long long ga = (unsigned long long)gt;
        unsigned n = 5u * (unsigned)G;          // elements (4B each)
        u32x4 g0;
        u32x8 g1;
        // D# group 0 (128b): count=1, lds_addr, global_addr, type=2
        g0[0] = 1u;                                         // count=1, no gather
        g0[1] = lds_off;                                    // lds_addr (bytes)
        g0[2] = (u32)(ga & 0xFFFFFFFFull);                  // global_addr[31:0]
        g0[3] = (u32)((ga >> 32) & 0x01FFFFFFull)           // global_addr[56:32]
                | (2u << 30);                               // type = 2 ("image")
        // D# group 1 (256b): data_size=4B, 1-D tensor/tile of n elements
        g1[0] = (2u << 16);                                 // data_size = 4 bytes
        g1[1] = (n << 16);                                  // tensor_dim0[15:0] @ bits 63:48
        g1[2] = (1u << 16);                                 // tensor_dim1 = 1   @ bits 95:80
        g1[3] = (n << 16);                                  // tile_dim0         @ bits 127:112
        g1[4] = 1u;                                         // tile_dim1 = 1, tile_dim2 = 0
        g1[5] = n;                                          // tensor_dim0_stride[31:0]
        g1[6] = 0u;
        g1[7] = 0u;
        // Pass the LDS pointer as an operand (unused in the template) so the shared
        // array escapes: alias analysis must then assume the asm writes sgt[], keeping
        // the downstream ds_load instructions alive.
        asm volatile("tensor_load_to_lds %0, %1"
                     :: "s"(g0), "s"(g1), "r"(sp) : "memory");
        __builtin_amdgcn_s_wait_tensorcnt(0);
    }
    __syncthreads();

    int a = blockIdx.x * 256 + threadIdx.x;
    if (a >= A) return;

    int fw = *pfw;                  // uniform scalar load
    (void)pfh;                      // fh only determines A (known from out_size)

    int k    = a % 9;
    int cell = a / 9;
    int cy   = cell / fw;
    int cx   = cell - cy * fw;
    float sx = (float)(cx * 8);
    float sy = (float)(cy * 8);

    float4 b = *(const float4*)&c_base[k][0];
    float ax1 = b.x + sx, ay1 = b.y + sy, ax2 = b.z + sx, ay2 = b.w + sy;
    float area_a = (ax2 - ax1 + 1.0f) * (ay2 - ay1 + 1.0f);

    // Division-free running max/argmax of IoU = inter/denom (denom > 0):
    // compare inter_g * best_den > best_inter * den_g (all terms >= 0, den > 0).
    // Strict '>' preserves jnp.argmax first-occurrence semantics.
    float bi = -1.0f, bd = 1.0f;
    int bidx = 0;
    #pragma unroll 4
    for (int g = 0; g < G; ++g) {
        const float* r = &sgt[5 * g];           // lane-uniform LDS broadcast
        float gx1 = r[0], gy1 = r[1], gx2 = r[2], gy2 = r[3];
        float iw = fminf(ax2, gx2) - fmaxf(ax1, gx1) + 1.0f;
        float ih = fminf(ay2, gy2) - fmaxf(ay1, gy1) + 1.0f;
        iw = fmaxf(iw, 0.0f);
        ih = fmaxf(ih, 0.0f);
        float inter  = iw * ih;
        float area_g = (gx2 - gx1 + 1.0f) * (gy2 - gy1 + 1.0f);
        float denom  = area_a + area_g - inter;
        if (inter * bd > bi * denom) { bi = inter; bd = denom; bidx = g; }
    }
    float mo = bi / bd;

    // Regression targets from the argmax GT row (applied to ALL anchors).
    const float* r = &sgt[5 * bidx];
    float gx1 = r[0], gy1 = r[1], gx2 = r[2], gy2 = r[3], gcls = r[4];
    float ew  = ax2 - ax1 + 1.0f;
    float eh  = ay2 - ay1 + 1.0f;
    float ecx = ax1 + 0.5f * ew;
    float ecy = ay1 + 0.5f * eh;
    float gw  = gx2 - gx1 + 1.0f;
    float gh  = gy2 - gy1 + 1.0f;
    float gcx = gx1 + 0.5f * gw;
    float gcy = gy1 + 0.5f * gh;
    float t0 = (gcx - ecx) / ew;
    float t1 = (gcy - ecy) / eh;
    float t2 = __logf(gw / ew);
    float t3 = __logf(gh / eh);

    float lab = -1.0f;
    if (mo < 0.4f)  lab = 0.0f;
    if (mo >= 0.5f) lab = gcls;   // labels==1 replaced by class of argmax box

    // Output layout: labels[A] ++ targets[A*4] ++ anchors[A*4].
    // Write-once data: use nontemporal stores (th:NT) to keep it out of cache.
    __builtin_nontemporal_store(lab, out + a);
    f32x4 tv = {t0, t1, t2, t3};
    f32x4 av = {ax1, ay1, ax2, ay2};
    __builtin_nontemporal_store(tv, (f32x4*)(out + (size_t)A) + a);
    __builtin_nontemporal_store(av, (f32x4*)(out + 5ull * (size_t)A) + a);
}

extern "C" void kernel_launch(void* const* d_in, const int* in_sizes, int n_in,
                              void* d_out, int out_size, void* d_ws, size_t ws_size,
                              hipStream_t stream) {
    const float* gt  = (const float*)d_in[0];   // (1, G, 5) float32
    const int*   pfw = (const int*)d_in[1];     // scalar
    const int*   pfh = (const int*)d_in[2];     // scalar
    float* out = (float*)d_out;

    int G = in_sizes[0] / 5;                    // 100
    int A = out_size / 9;                       // labels + 4*targets + 4*anchors
    int blocks = (A + 255) / 256;

    anchor_target_kernel<<<dim3(blocks), dim3(256), 0, stream>>>(gt, pfw, pfh, out, A, G);
}